// UFGConv_90744069030479
// MI455X (gfx1250) — compile-verified
//
#include <hip/hip_runtime.h>
#include <math.h>

#define FD 256  // F_OUT

typedef __attribute__((ext_vector_type(2))) float v2f;
typedef __attribute__((ext_vector_type(8))) float v8f;

// ---------------- elementwise helpers ----------------

__global__ __launch_bounds__(256) void k_fill(float* __restrict__ p, float v, long n) {
  long i = (long)blockIdx.x * blockDim.x + threadIdx.x;
  if (i < n) p[i] = v;
}

// y[node][f] = b[f]  (initializes an accumulator with its bias)
__global__ __launch_bounds__(256) void k_rows_bias(float* __restrict__ y,
                                                   const float* __restrict__ b, int nodes) {
  long i = (long)blockIdx.x * blockDim.x + threadIdx.x;
  long n = (long)nodes * FD;
  if (i < n) y[i] = b[i & (FD - 1)];
}

__global__ __launch_bounds__(256) void k_elu(float* __restrict__ y, long n) {
  long i = (long)blockIdx.x * blockDim.x + threadIdx.x;
  if (i < n) {
    float v = y[i];
    y[i] = v > 0.f ? v : (__expf(v) - 1.f);
  }
}

// ---------------- dense GEMM via V_WMMA_F32_16X16X4_F32 ----------------
// C[M,Ncols] = A[M,K] @ B[K,Ncols].  Each wave computes a 16(M) x 64(N)
// strip with 4 accumulators; one A fragment per K-step feeds 4 WMMAs
// (independent D registers -> no WMMA->WMMA RAW hazard, good pipelining).
// fp32 fragment layout (ISA 7.12.2): lanes 0-15 hold rows 0..15 with
// (K0,K1) in the 2 fragment VGPRs, lanes 16-31 the same rows with (K2,K3).
// 4 waves/block cover 16 x 256.  Requires Ncols % 64 == 0 (true here: 256).

__global__ __launch_bounds__(128)
void k_gemm_wmma(const float* __restrict__ A, const float* __restrict__ B,
                 float* __restrict__ C, int M, int K, int Ncols) {
  const int lane = threadIdx.x & 31;
  const int wave = threadIdx.x >> 5;
  const int r    = lane & 15;   // row (A) / col (B,C) within tile
  const int kh   = lane >> 4;   // K-half selector
  const int n0   = blockIdx.x * 256 + wave * 64;
  const int m0   = blockIdx.y * 16;
  if (n0 >= Ncols || m0 >= M) return;   // wave-uniform: EXEC stays all-1s

  int arow = m0 + r;
  if (arow >= M) arow = M - 1;          // clamp (keeps EXEC full)
  const float* Arow = A + (size_t)arow * K;
  const float* Bp   = B + n0 + r;

  v8f acc[4];
#pragma unroll
  for (int t = 0; t < 4; ++t) acc[t] = (v8f){0.f, 0.f, 0.f, 0.f, 0.f, 0.f, 0.f, 0.f};

  for (int k0 = 0; k0 < K; k0 += 4) {
    const int ka = k0 + 2 * kh;
    v2f a;
    a.x = Arow[ka];
    a.y = Arow[ka + 1];
    const float* B0 = Bp + (size_t)ka * Ncols;        // row ka
    const float* B1 = Bp + (size_t)(ka + 1) * Ncols;  // row ka+1
#pragma unroll
    for (int t = 0; t < 4; ++t) {
      v2f b;
      b.x = B0[16 * t];
      b.y = B1[16 * t];
      acc[t] = __builtin_amdgcn_wmma_f32_16x16x4_f32(false, a, false, b,
                                                     (short)0, acc[t], false, false);
    }
  }

#pragma unroll
  for (int t = 0; t < 4; ++t) {
#pragma unroll
    for (int i = 0; i < 8; ++i) {
      int orow = m0 + i + 8 * kh;
      if (orow < M) C[(size_t)orow * Ncols + n0 + 16 * t + r] = acc[t][i];
    }
  }
}

// ---------------- edge-parallel SpMM: x[row] += val * h[col] ----------------

__global__ __launch_bounds__(256)
void k_spmm(const int* __restrict__ row, const int* __restrict__ col,
            const float* __restrict__ val, const float* __restrict__ h,
            float* __restrict__ x, int nnz) {
  const int lane = threadIdx.x & 31;
  const int wid  = (blockIdx.x * blockDim.x + threadIdx.x) >> 5;
  const int nw   = (gridDim.x * blockDim.x) >> 5;
  for (int e = wid; e < nnz; e += nw) {
    int en = e + nw;
    if (en < nnz)  // pull next gathered row toward L2/L0 (global_prefetch_b8)
      __builtin_prefetch(h + (size_t)col[en] * FD + lane * 8, 0, 3);
    int r = row[e], c = col[e];
    float v = val[e];
    const float4* hs = (const float4*)(h + (size_t)c * FD);
    float* xr = x + (size_t)r * FD;
#pragma unroll
    for (int i = 0; i < 2; ++i) {
      float4 hv = hs[lane + 32 * i];
      int base = (lane + 32 * i) * 4;
      atomicAdd(xr + base + 0, v * hv.x);
      atomicAdd(xr + base + 1, v * hv.y);
      atomicAdd(xr + base + 2, v * hv.z);
      atomicAdd(xr + base + 3, v * hv.w);
    }
  }
}

// ------- transposed SpMM: out[col] += val * (rowscale[row]) * src[row] -------

__global__ __launch_bounds__(256)
void k_spmmt(const int* __restrict__ row, const int* __restrict__ col,
             const float* __restrict__ val, const float* __restrict__ src,
             const float* __restrict__ rowscale, float* __restrict__ out, int nnz) {
  const int lane = threadIdx.x & 31;
  const int wid  = (blockIdx.x * blockDim.x + threadIdx.x) >> 5;
  const int nw   = (gridDim.x * blockDim.x) >> 5;
  for (int e = wid; e < nnz; e += nw) {
    int en = e + nw;
    if (en < nnz)
      __builtin_prefetch(src + (size_t)row[en] * FD + lane * 8, 0, 3);
    int r = row[e], c = col[e];
    float v = val[e];
    if (rowscale) v *= rowscale[r];
    const float4* ss = (const float4*)(src + (size_t)r * FD);
    float* oc = out + (size_t)c * FD;
#pragma unroll
    for (int i = 0; i < 2; ++i) {
      float4 sv = ss[lane + 32 * i];
      int base = (lane + 32 * i) * 4;
      atomicAdd(oc + base + 0, v * sv.x);
      atomicAdd(oc + base + 1, v * sv.y);
      atomicAdd(oc + base + 2, v * sv.z);
      atomicAdd(oc + base + 3, v * sv.w);
    }
  }
}

// ---------------- GAT pieces ----------------

// s_src[n] = g[n] . a_src ; s_dst[n] = g[n] . a_dst  (one wave per node)
__global__ __launch_bounds__(256)
void k_gat_proj(const float* __restrict__ g, const float* __restrict__ asrc,
                const float* __restrict__ adst, float* __restrict__ ssrc,
                float* __restrict__ sdst, int nodes) {
  int wave = blockIdx.x * (blockDim.x >> 5) + (threadIdx.x >> 5);
  int lane = threadIdx.x & 31;
  if (wave >= nodes) return;
  const float* rowp = g + (size_t)wave * FD;
  float ss = 0.f, sd = 0.f;
#pragma unroll
  for (int j = 0; j < FD / 32; ++j) {
    float v = rowp[lane + 32 * j];
    ss = fmaf(v, asrc[lane + 32 * j], ss);
    sd = fmaf(v, adst[lane + 32 * j], sd);
  }
#pragma unroll
  for (int off = 16; off > 0; off >>= 1) {
    ss += __shfl_xor(ss, off, 32);
    sd += __shfl_xor(sd, off, 32);
  }
  if (lane == 0) { ssrc[wave] = ss; sdst[wave] = sd; }
}

__device__ __forceinline__ void atomic_max_f32(float* addr, float v) {
  // works for mixed signs when initialized to -inf
  if (v >= 0.f) atomicMax((int*)addr, __float_as_int(v));
  else          atomicMin((unsigned int*)addr, (unsigned int)__float_as_int(v));
}

// alpha = leaky_relu(ssrc[src]+sdst[dst]); segment-max over dst
__global__ __launch_bounds__(256)
void k_gat_alpha(const int* __restrict__ src, const int* __restrict__ dst,
                 const float* __restrict__ ssrc, const float* __restrict__ sdst,
                 float* __restrict__ abuf, float* __restrict__ amax, int E, int nodes) {
  int e = blockIdx.x * blockDim.x + threadIdx.x;
  int tot = E + nodes;                 // self-loops appended
  if (e >= tot) return;
  int s, d;
  if (e < E) { s = src[e]; d = dst[e]; } else { s = e - E; d = s; }
  float a = ssrc[s] + sdst[d];
  a = a > 0.f ? a : 0.2f * a;          // NEG_SLOPE = 0.2
  abuf[e] = a;
  atomic_max_f32(amax + d, a);
}

// e = exp(alpha - amax[dst]); denom[dst] += e
__global__ __launch_bounds__(256)
void k_gat_expsum(const int* __restrict__ dst, const float* __restrict__ amax,
                  float* __restrict__ abuf, float* __restrict__ denom, int E, int nodes) {
  int e = blockIdx.x * blockDim.x + threadIdx.x;
  int tot = E + nodes;
  if (e >= tot) return;
  int d = (e < E) ? dst[e] : (e - E);
  float ee = __expf(abuf[e] - amax[d]);
  abuf[e] = ee;
  atomicAdd(denom + d, ee);
}

// y[dst] += (e/denom[dst]) * g[src]   (one wave per edge)
__global__ __launch_bounds__(256)
void k_gat_scatter(const int* __restrict__ src, const int* __restrict__ dst,
                   const float* __restrict__ ebuf, const float* __restrict__ denom,
                   const float* __restrict__ g, float* __restrict__ y, int E, int nodes) {
  const int lane = threadIdx.x & 31;
  const int wid  = (blockIdx.x * blockDim.x + threadIdx.x) >> 5;
  const int nw   = (gridDim.x * blockDim.x) >> 5;
  const int tot  = E + nodes;
  for (int e = wid; e < tot; e += nw) {
    int s, d;
    if (e < E) { s = src[e]; d = dst[e]; } else { s = e - E; d = s; }
    float attn = ebuf[e] / (denom[d] + 1e-16f);
    const float4* gs = (const float4*)(g + (size_t)s * FD);
    float* yd = y + (size_t)d * FD;
#pragma unroll
    for (int i = 0; i < 2; ++i) {
      float4 v = gs[lane + 32 * i];
      int base = (lane + 32 * i) * 4;
      atomicAdd(yd + base + 0, attn * v.x);
      atomicAdd(yd + base + 1, attn * v.y);
      atomicAdd(yd + base + 2, attn * v.z);
      atomicAdd(yd + base + 3, attn * v.w);
    }
  }
}

// ---------------- orchestration ----------------

extern "C" void kernel_launch(void* const* d_in, const int* in_sizes, int n_in,
                              void* d_out, int out_size, void* d_ws, size_t ws_size,
                              hipStream_t stream) {
  (void)n_in; (void)out_size; (void)ws_size;

  const float* x    = (const float*)d_in[0];
  const int*   ei   = (const int*)  d_in[1];
  const int E = in_sizes[1] / 2;
  const int* esrc = ei;
  const int* edst = ei + E;
  const int*   d1r = (const int*)  d_in[2];
  const int*   d1c = (const int*)  d_in[3];
  const float* d1v = (const float*)d_in[4];
  const int*   d2r = (const int*)  d_in[5];
  const int*   d2c = (const int*)  d_in[6];
  const float* d2v = (const float*)d_in[7];
  const int*   d3r = (const int*)  d_in[8];
  const int*   d3c = (const int*)  d_in[9];
  const float* d3v = (const float*)d_in[10];
  const float* W    = (const float*)d_in[11];
  const float* bias = (const float*)d_in[12];
  const float* filt = (const float*)d_in[13];
  const float* w2   = (const float*)d_in[14];
  const float* a2s  = (const float*)d_in[15];
  const float* a2d  = (const float*)d_in[16];
  const float* b2   = (const float*)d_in[17];
  const float* w3   = (const float*)d_in[18];
  const float* a3s  = (const float*)d_in[19];
  const float* a3d  = (const float*)d_in[20];
  const float* b3   = (const float*)d_in[21];

  const int Nn   = in_sizes[13];           // filter1 is [N,1]
  const int Fin  = in_sizes[0] / Nn;       // 512
  const int nnz1 = in_sizes[2], nnz2 = in_sizes[5], nnz3 = in_sizes[8];
  const int Etot = E + Nn;

  // workspace carve (floats)
  float* ws   = (float*)d_ws;
  const size_t S = (size_t)Nn * FD;
  float* bufH = ws;              // h, then g2, then g3
  float* bufA = ws + S;          // x1, then y2/y3 (GAT outputs)
  float* bufB = ws + 2 * S;      // x2
  float* bufC = ws + 3 * S;      // x3
  float* ebuf = ws + 4 * S;      // per-edge alpha / exp   [Etot]
  float* ssrc = ebuf + Etot;
  float* sdst = ssrc + Nn;
  float* amax = sdst + Nn;
  float* denm = amax + Nn;
  float* out  = (float*)d_out;

  const long elems = (long)Nn * FD;
  const int  fillg = (int)((elems + 255) / 256);
  const dim3 gemm_grid((FD + 255) / 256, (Nn + 15) / 16);  // 4 waves/block cover N=256
  const int  SPMM_BLOCKS = 4096;           // grid-stride, 8 waves/block
  const float NEG_INF = -__builtin_huge_valf();

  // 1) h = x @ W
  k_gemm_wmma<<<gemm_grid, 128, 0, stream>>>(x, W, bufH, Nn, Fin, FD);

  // 2) x1,x2,x3 = 0 (contiguous), then three SpMMs
  k_fill<<<(int)((3 * S + 255) / 256), 256, 0, stream>>>(bufA, 0.f, (long)(3 * S));
  k_spmm<<<SPMM_BLOCKS, 256, 0, stream>>>(d1r, d1c, d1v, bufH, bufA, nnz1);
  k_spmm<<<SPMM_BLOCKS, 256, 0, stream>>>(d2r, d2c, d2v, bufH, bufB, nnz2);
  k_spmm<<<SPMM_BLOCKS, 256, 0, stream>>>(d3r, d3c, d3v, bufH, bufC, nnz3);

  // 3) out = bias (broadcast), then out += d1^T @ (filter1 * x1)
  k_rows_bias<<<fillg, 256, 0, stream>>>(out, bias, Nn);
  k_spmmt<<<SPMM_BLOCKS, 256, 0, stream>>>(d1r, d1c, d1v, bufA, filt, out, nnz1);

  // 4) GAT on x2  (g2 reuses bufH; y2 reuses bufA)
  k_gemm_wmma<<<gemm_grid, 128, 0, stream>>>(bufB, w2, bufH, Nn, FD, FD);
  k_gat_proj<<<(Nn + 7) / 8, 256, 0, stream>>>(bufH, a2s, a2d, ssrc, sdst, Nn);
  k_fill<<<(Nn + 255) / 256, 256, 0, stream>>>(amax, NEG_INF, Nn);
  k_fill<<<(Nn + 255) / 256, 256, 0, stream>>>(denm, 0.f, Nn);
  k_gat_alpha<<<(Etot + 255) / 256, 256, 0, stream>>>(esrc, edst, ssrc, sdst, ebuf, amax, E, Nn);
  k_gat_expsum<<<(Etot + 255) / 256, 256, 0, stream>>>(edst, amax, ebuf, denm, E, Nn);
  k_rows_bias<<<fillg, 256, 0, stream>>>(bufA, b2, Nn);
  k_gat_scatter<<<SPMM_BLOCKS, 256, 0, stream>>>(esrc, edst, ebuf, denm, bufH, bufA, E, Nn);
  k_elu<<<fillg, 256, 0, stream>>>(bufA, elems);
  k_spmmt<<<SPMM_BLOCKS, 256, 0, stream>>>(d2r, d2c, d2v, bufA, nullptr, out, nnz2);

  // 5) GAT on x3  (g3 reuses bufH; y3 reuses bufA)
  k_gemm_wmma<<<gemm_grid, 128, 0, stream>>>(bufC, w3, bufH, Nn, FD, FD);
  k_gat_proj<<<(Nn + 7) / 8, 256, 0, stream>>>(bufH, a3s, a3d, ssrc, sdst, Nn);
  k_fill<<<(Nn + 255) / 256, 256, 0, stream>>>(amax, NEG_INF, Nn);
  k_fill<<<(Nn + 255) / 256, 256, 0, stream>>>(denm, 0.f, Nn);
  k_gat_alpha<<<(Etot + 255) / 256, 256, 0, stream>>>(esrc, edst, ssrc, sdst, ebuf, amax, E, Nn);
  k_gat_expsum<<<(Etot + 255) / 256, 256, 0, stream>>>(edst, amax, ebuf, denm, E, Nn);
  k_rows_bias<<<fillg, 256, 0, stream>>>(bufA, b3, Nn);
  k_gat_scatter<<<SPMM_BLOCKS, 256, 0, stream>>>(esrc, edst, ebuf, denm, bufH, bufA, E, Nn);
  k_elu<<<fillg, 256, 0, stream>>>(bufA, elems);
  k_spmmt<<<SPMM_BLOCKS, 256, 0, stream>>>(d3r, d3c, d3v, bufA, nullptr, out, nnz3);
}